// scale_and_CDF_10728828305564
// MI455X (gfx1250) — compile-verified
//
#include <hip/hip_runtime.h>

typedef __attribute__((ext_vector_type(2))) float v2f;
typedef __attribute__((ext_vector_type(8))) float v8f;
typedef __attribute__((ext_vector_type(4))) unsigned int u32x4;
typedef __attribute__((ext_vector_type(4))) int i32x4;
typedef __attribute__((ext_vector_type(8))) int i32x8;

#define BATCH   524288
#define NDIM    64
#define NBINS   32
#define BOUNDF  50.0f
#define RGEO    1.2f
#define LN2F    0.69314718055994531f

// ---- workspace layout (floats) ----
#define NBLK_A   512
#define PART_OFF 0                    // [NBLK_A][128]  (sum[64], sumsq[64] per block)
#define TAB_OFF  (NBLK_A * 128)       // 65536
// table region (relative to TAB_OFF); pdf/fref use pitch 65 -> k maps to LDS bank
#define T_SHIFT  0
#define T_SCL01  64                   // exp(s)/100
#define T_PDF    128                  // [33][65]
#define T_FREF   (128 + 33 * 65)      // 2273, [32][65]
#define T_MESH   (T_FREF + 32 * 65)   // 4353, [33]
#define T_ELMT   (T_MESH + 33)        // 4386, [32]
#define T_IELMT  (T_ELMT + 32)        // 4418, [32]  1/elmt
#define T_SUMS   (T_IELMT + 32)       // 4450
#define T_X1L    (T_SUMS + 1)         // 4451
#define TAB_LEN  (T_X1L + 1)          // 4452

// ------------------------------------------------------------------
// Pass 1: per-dim partial sums / sums-of-squares (deterministic, no atomics)
// ------------------------------------------------------------------
__global__ void nf_stats(const float* __restrict__ x, float* __restrict__ ws) {
    __shared__ float ss[256];
    __shared__ float sq[256];
    const int tid = threadIdx.x;
    const int d   = tid & 63;
    const int rg  = tid >> 6;             // 0..3
    const size_t base = (size_t)blockIdx.x * 1024;
    float s = 0.f, q = 0.f;
    for (int i = 0; i < 256; ++i) {
        size_t row = base + (size_t)rg + (size_t)i * 4;
        if ((i & 7) == 0)
            __builtin_prefetch(x + row * NDIM + d + 16384, 0, 0);  // global_prefetch_b8
        float v = x[row * NDIM + d];
        s += v;
        q += v * v;
    }
    ss[tid] = s; sq[tid] = q;
    __syncthreads();
    if (tid < 64) {
        float S = ss[tid] + ss[tid + 64] + ss[tid + 128] + ss[tid + 192];
        float Q = sq[tid] + sq[tid + 64] + sq[tid + 128] + sq[tid + 192];
        float* pb = ws + PART_OFF + blockIdx.x * 128;
        pb[tid]      = S;
        pb[64 + tid] = Q;
    }
}

// ------------------------------------------------------------------
// Pass 2: single tiny block — finalize actnorm params + pdf/CDF tables
// ------------------------------------------------------------------
__global__ void nf_prep(const float* __restrict__ p, const float* __restrict__ b,
                        const float* __restrict__ logs, float* __restrict__ ws) {
    __shared__ float mesh[33];
    __shared__ float elmt[32];
    __shared__ float sval[64];
    const int t = threadIdx.x;  // 64 threads

    if (t < 33) {
        double x1L_raw = 50.0 * (1.2 - 1.0) / (pow(1.2, 16.0) - 1.0);
        double idx = (double)t - 16.0;
        double xr  = (1.0 - pow(1.2, fabs(idx))) / (1.0 - 1.2);
        xr = (idx >= 0.0) ? x1L_raw * xr : -x1L_raw * xr;
        xr = (xr + 50.0) / 100.0;
        float mv = (t == 0) ? 0.f : ((t == 32) ? 1.f : (float)xr);
        mesh[t] = mv;
        ws[TAB_OFF + T_MESH + t] = mv;
        if (t == 0) ws[TAB_OFF + T_X1L] = (float)(x1L_raw / 100.0);
    }
    __syncthreads();
    if (t < 32) {
        float e = mesh[t + 1] - mesh[t];
        elmt[t] = e;
        ws[TAB_OFF + T_ELMT + t]  = e;
        ws[TAB_OFF + T_IELMT + t] = 1.0f / e;
    }
    __syncthreads();

    // per-dimension work (t = dim, 0..63)
    float S = 0.f, Q = 0.f;
    for (int i = 0; i < NBLK_A; ++i) {
        S += ws[PART_OFF + i * 128 + t];
        Q += ws[PART_OFF + i * 128 + 64 + t];
    }
    const float mean = S * (1.0f / (float)BATCH);
    const float var  = Q * (1.0f / (float)BATCH) - mean * mean;
    const float logs_init = logf(1.0f / (sqrtf(var) + 1e-6f)) * (1.0f / 3.0f);
    float s = logs[t] + logs_init;
    s = fminf(5.0f, fmaxf(-5.0f, s));
    sval[t] = s;
    ws[TAB_OFF + T_SHIFT + t] = b[t] - mean;
    ws[TAB_OFF + T_SCL01 + t] = expf(s) * (0.5f / BOUNDF);   // exp(s)/100

    float denom = 0.f;
    for (int j = 0; j < NBINS - 1; ++j)
        denom += expf(p[j * NDIM + t]) * 0.5f * (elmt[j] + elmt[j + 1]);
    const float norm = (1.0f - elmt[0]) / denom;

    ws[TAB_OFF + T_PDF + 0 * 65 + t] = 1.0f;
    float run = 0.f, pj = 1.0f;
    for (int j = 0; j < NBINS; ++j) {
        float pn = (j == NBINS - 1) ? 1.0f : norm * expf(p[j * NDIM + t]);
        ws[TAB_OFF + T_PDF + (j + 1) * 65 + t] = pn;
        ws[TAB_OFF + T_FREF + j * 65 + t]      = run;   // F_ref[j] = sum_{i<j} cell[i]
        run += 0.5f * (pj + pn) * elmt[j];
        pj = pn;
    }
    if (t == 0) {   // init pitch-pad columns (ws is poisoned 0xAA by harness)
        for (int j = 0; j < 33; ++j) ws[TAB_OFF + T_PDF + j * 65 + 64] = 0.f;
        for (int j = 0; j < 32; ++j) ws[TAB_OFF + T_FREF + j * 65 + 64] = 0.f;
    }
    __syncthreads();
    if (t == 0) {
        float acc = 0.f;
        for (int i = 0; i < NDIM; ++i) acc += sval[i];  // fixed order -> deterministic
        ws[TAB_OFF + T_SUMS] = acc;
    }
}

// ------------------------------------------------------------------
// TDM helper: async-load TAB_LEN floats (1-row 2D tile) from ws into LDS.
// D# built per CDNA5 ISA 8.3/8.4: count=1, type=2, data_size=4B,
// tensor_dim0 = tile_dim0 = TAB_LEN, tensor_dim1 = tile_dim1 = 1.
// ------------------------------------------------------------------
__device__ __forceinline__ void tdm_load_tab(const float* gsrc, float* lds_tab) {
    const unsigned lds_addr = (unsigned)(unsigned long long)(uintptr_t)lds_tab;
    const unsigned long long ga = (unsigned long long)(uintptr_t)gsrc;

    u32x4 g0;
    g0[0] = 1u;                                   // count=1, is_restore=0, gather=0
    g0[1] = lds_addr;                             // lds_addr
    g0[2] = (unsigned)ga;                         // global_addr[31:0]
    g0[3] = (unsigned)(ga >> 32) | (2u << 30);    // global_addr[56:32] | type=2

    i32x8 g1;
    g1[0] = 0x00020000;                           // wg_mask=0, data_size=2 (4B)
    g1[1] = (int)(TAB_LEN << 16);                 // tensor_dim0[15:0] in [31:16]
    g1[2] = 0x00010000;                           // tensor_dim0 hi=0 | tensor_dim1=1
    g1[3] = (int)(TAB_LEN << 16);                 // tensor_dim1 hi=0 | tile_dim0
    g1[4] = 0x00000001;                           // tile_dim1=1, tile_dim2=0
    g1[5] = (int)TAB_LEN;                         // tensor_dim0_stride[31:0]
    g1[6] = (int)(TAB_LEN << 16);                 // dim0_stride hi=0 | dim1_stride lo
    g1[7] = 0;                                    // dim1_stride hi

    i32x4 gz = {0, 0, 0, 0};
#if __has_include(<hip/amd_detail/amd_gfx1250_TDM.h>)
    i32x8 gz8 = {0, 0, 0, 0, 0, 0, 0, 0};
    __builtin_amdgcn_tensor_load_to_lds(g0, g1, gz, gz, gz8, 0);   // clang-23 arity
#else
    __builtin_amdgcn_tensor_load_to_lds(g0, g1, gz, gz, 0);        // ROCm 7.2 arity
#endif
}

// ------------------------------------------------------------------
// Pass 3: streaming CDF transform; per-row logdet via V_WMMA_F32_16X16X4_F32
// Lane layout matches ISA 16x4 f32 A-matrix: lane&15 = row M, lane>>4 picks K pair.
// B = ones => D[m,n] = sum_k A[m,k] + C (row sums), C-chained over 16 K-chunks.
// logdet accumulated in log2 units; one *ln2 at the end.
// ------------------------------------------------------------------
__global__ void nf_main(const float* __restrict__ x, const float* __restrict__ ld_in,
                        const float* __restrict__ ws, float* __restrict__ y_out,
                        float* __restrict__ ld_out) {
    __shared__ float tab[TAB_LEN];
    const int tid = threadIdx.x;
    if (tid < 32) {                               // wave 0 only: one TDM op per block
        tdm_load_tab(ws + TAB_OFF, tab);
        __builtin_amdgcn_s_wait_tensorcnt(0);
    }
    __syncthreads();

    const float* s_shift = tab + T_SHIFT;
    const float* s_scl01 = tab + T_SCL01;
    const float* s_pdf   = tab + T_PDF;
    const float* s_fref  = tab + T_FREF;
    const float* s_mesh  = tab + T_MESH;
    const float* s_ielmt = tab + T_IELMT;
    const float  sum_s   = tab[T_SUMS];
    const float  x1l     = tab[T_X1L];
    const float  c1       = (RGEO - 1.0f) / x1l;
    const float  invL2R   = 1.0f / __log2f(RGEO);   // 1/log2(1.2)

    const int wid  = tid >> 5;
    const int lane = tid & 31;
    const int half = lane >> 4;        // selects K pair {0,1} vs {2,3}
    const int l16  = lane & 15;        // row within 16-row tile
    const int rowBase = blockIdx.x * 128 + wid * 16;
    const int row     = rowBase + l16;
    const size_t xbase = (size_t)row * NDIM;

    v8f acc = {0.f, 0.f, 0.f, 0.f, 0.f, 0.f, 0.f, 0.f};
    v2f ones; ones[0] = 1.0f; ones[1] = 1.0f;

#pragma unroll
    for (int c = 0; c < 16; ++c) {
        const int d0 = c * 4 + half * 2;
        float2 xv = *reinterpret_cast<const float2*>(x + xbase + d0);
        float xin[2] = {xv.x, xv.y};
        float ll[2], yo[2];
#pragma unroll
        for (int e = 0; e < 2; ++e) {
            const int d = d0 + e;
            float t = (xin[e] + s_shift[d]) * s_scl01[d];   // u - 0.5
            float u = t + 0.5f;
            float mval = floorf(__log2f(fmaf(fabsf(t), c1, 1.0f)) * invL2R);
            int   k    = (t >= 0.f) ? (16 + (int)mval) : (15 - (int)mval);
            bool  cover = ((unsigned)k) < (unsigned)NBINS;
            int   kk = min(max(k, 0), NBINS - 1);
            // branchless table reads (kk clamped => always safe)
            float v1 = s_pdf[kk * 65 + d];
            float v2 = s_pdf[(kk + 1) * 65 + d];
            float Fp = s_fref[kk * 65 + d];
            float xm = u - s_mesh[kk];
            float slope = (v2 - v1) * s_ielmt[kk];
            float ycov = fmaf(xm, fmaf(xm * 0.5f, slope, v1), Fp);
            float y    = cover ? ycov : u;
            float dld  = cover ? fmaf(xm, slope, v1) : 1.0f;  // log2(1)=0
            ll[e] = __log2f(dld);
            yo[e] = fmaf(y, 2.0f * BOUNDF, -BOUNDF);
        }
        *reinterpret_cast<float2*>(y_out + xbase + d0) = make_float2(yo[0], yo[1]);

        v2f a; a[0] = ll[0]; a[1] = ll[1];
        // D = A(16x4 of log2-dets) x B(ones) + C  -> row sums accumulate in D
        acc = __builtin_amdgcn_wmma_f32_16x16x4_f32(
            false, a, false, ones, (short)0, acc, false, false);
    }

    // D layout: VGPR j -> (M=j, lanes 0-15) / (M=j+8, lanes 16-31); rows replicated over N.
    if (l16 == 0) {
        const int rb = rowBase + half * 8;
#pragma unroll
        for (int j = 0; j < 8; ++j) {
            const int r = rb + j;
            ld_out[r] = ld_in[r] + sum_s + acc[j] * LN2F;
        }
    }
}

// ------------------------------------------------------------------
extern "C" void kernel_launch(void* const* d_in, const int* in_sizes, int n_in,
                              void* d_out, int out_size, void* d_ws, size_t ws_size,
                              hipStream_t stream) {
    const float* x    = (const float*)d_in[0];
    const float* ld   = (const float*)d_in[1];
    const float* p    = (const float*)d_in[2];
    const float* b    = (const float*)d_in[3];
    const float* logs = (const float*)d_in[4];
    float* ws    = (float*)d_ws;
    float* y_out = (float*)d_out;
    float* ldout = y_out + (size_t)BATCH * NDIM;

    nf_stats<<<NBLK_A, 256, 0, stream>>>(x, ws);
    nf_prep <<<1, 64, 0, stream>>>(p, b, logs, ws);
    nf_main <<<BATCH / 128, 256, 0, stream>>>(x, ld, ws, y_out, ldout);
}